// AffinityPredictor_28243704939144
// MI455X (gfx1250) — compile-verified
//
#include <hip/hip_runtime.h>
#include <hip/hip_bf16.h>

// GCN: h1 = relu(GCNconv(x,W1,b1)); h2 = relu(GCNconv(h1,W2,b2)); mean-pool; FC.
// Strategy: edge lists stream from HBM (23.3 TB/s), node features (25.6MB) and
// accumulators live in the 192MB L2, GEMMs use V_WMMA_F32_16X16X4_F32.

#define IN_DIM   20
#define NODE_DIM 64

typedef float v2f __attribute__((ext_vector_type(2)));
typedef float v8f __attribute__((ext_vector_type(8)));

// ---------------- degree / norm ----------------
__global__ void k_degree(const int* __restrict__ dst, float* __restrict__ deg, int E) {
    int i = blockIdx.x * blockDim.x + threadIdx.x;
    if (i < E) atomicAdd(&deg[dst[i]], 1.0f);
}

__global__ void k_dinv(float* __restrict__ deg, int N) {
    int i = blockIdx.x * blockDim.x + threadIdx.x;
    if (i < N) deg[i] = __frsqrt_rn(deg[i] + 1.0f);   // +1 = self loop; always > 0
}

// ---------------- WMMA GEMM: Y[N x 64] = X[N x K] @ W[K x 64] ----------------
// Block = 128 threads = 4 waves; wave w computes the 16x16 tile at cols [16w,16w+16)
// of row tile blockIdx.x. K is 20 or 64 (multiple of 4). N is a multiple of 16.
__global__ void k_gemm_wmma(const float* __restrict__ X, const float* __restrict__ W,
                            float* __restrict__ Y, int N, int K) {
    const int wave = threadIdx.x >> 5;
    const int lane = threadIdx.x & 31;
    const int lm   = lane & 15;   // M (or N) index within tile
    const int lh   = lane >> 4;   // half-wave selector
    const int row0 = blockIdx.x * 16;
    const int col0 = wave * 16;

    const float* __restrict__ xrow = X + (size_t)(row0 + lm) * K;
    v8f c = {0.f, 0.f, 0.f, 0.f, 0.f, 0.f, 0.f, 0.f};

    for (int k0 = 0; k0 < K; k0 += 4) {
        // A 16x4: vgpr j, lane l -> A[l%16][k0 + j + 2*(l/16)]
        v2f a, b;
        a.x = xrow[k0 + 2 * lh + 0];
        a.y = xrow[k0 + 2 * lh + 1];
        // B 4x16: vgpr j, lane l -> B[k0 + j + 2*(l/16)][col0 + l%16]
        b.x = W[(size_t)(k0 + 2 * lh + 0) * NODE_DIM + col0 + lm];
        b.y = W[(size_t)(k0 + 2 * lh + 1) * NODE_DIM + col0 + lm];
        c = __builtin_amdgcn_wmma_f32_16x16x4_f32(
                /*neg_a=*/false, a, /*neg_b=*/false, b,
                /*c_mod=*/(short)0, c, /*reuse_a=*/false, /*reuse_b=*/false);
    }

    // C/D layout: vgpr i, lane l -> D[i + 8*(l/16)][l%16]
    float* __restrict__ yout = Y + (size_t)row0 * NODE_DIM + col0;
#pragma unroll
    for (int i = 0; i < 8; ++i) {
        int m = i + 8 * lh;
        yout[(size_t)m * NODE_DIM + lm] = c[i];
    }
}

// ---------------- edge scatter: agg[dst] += xw[src] * dinv[src]*dinv[dst] ----------------
// 16 threads per edge, each moves a float4 (B128 gather, 4x global_atomic_add_f32).
__global__ void k_scatter(const int* __restrict__ src, const int* __restrict__ dst,
                          const float* __restrict__ dinv, const float* __restrict__ xw,
                          float* __restrict__ agg, int E) {
    long long idx = (long long)blockIdx.x * blockDim.x + threadIdx.x;
    long long e   = idx >> 4;
    if (e >= E) return;
    int c = (int)(idx & 15) * 4;
    int s = src[e], d = dst[e];
    float nrm = dinv[s] * dinv[d];
    const float4 v = *(const float4*)(xw + (size_t)s * NODE_DIM + c);
    float* o = agg + (size_t)d * NODE_DIM + c;
    atomicAdd(o + 0, v.x * nrm);
    atomicAdd(o + 1, v.y * nrm);
    atomicAdd(o + 2, v.z * nrm);
    atomicAdd(o + 3, v.w * nrm);
}

// ---------------- fused self-loop + bias + relu (in place on agg) ----------------
__global__ void k_self_bias_relu(float* __restrict__ agg, const float* __restrict__ xw,
                                 const float* __restrict__ dinv, const float* __restrict__ b,
                                 int N) {
    long long idx = (long long)blockIdx.x * blockDim.x + threadIdx.x;
    if (idx >= (long long)N * NODE_DIM) return;
    int i = (int)(idx >> 6);
    int d = (int)(idx & 63);
    float di = dinv[i];
    float v = agg[idx] + xw[idx] * di * di + b[d];
    agg[idx] = v > 0.0f ? v : 0.0f;
}

// ---------------- fused self-loop + bias + relu + mean-pool atomics ----------------
__global__ void k_relu_pool(const float* __restrict__ agg, const float* __restrict__ xw,
                            const float* __restrict__ dinv, const float* __restrict__ b,
                            const int* __restrict__ batch, float* __restrict__ pooled,
                            float* __restrict__ cnt, int N) {
    long long idx = (long long)blockIdx.x * blockDim.x + threadIdx.x;
    if (idx >= (long long)N * NODE_DIM) return;
    int i = (int)(idx >> 6);
    int d = (int)(idx & 63);
    float di = dinv[i];
    float v = agg[idx] + xw[idx] * di * di + b[d];
    v = v > 0.0f ? v : 0.0f;
    int g = batch[i];
    atomicAdd(&pooled[(size_t)g * NODE_DIM + d], v);
    if (d == 0) atomicAdd(&cnt[g], 1.0f);
}

// ---------------- final FC: out[g] = dot(pooled[g]/max(cnt,1), Wfc) + bfc ----------------
// One wave32 per graph; lane l covers dims l and l+32; shuffle-tree reduce.
__global__ void k_final(const float* __restrict__ pooled, const float* __restrict__ cnt,
                        const float* __restrict__ Wfc, const float* __restrict__ bfc,
                        float* __restrict__ out, int G) {
    int g    = blockIdx.x * (blockDim.x >> 5) + (threadIdx.x >> 5);
    int lane = threadIdx.x & 31;
    if (g >= G) return;
    const float* p = pooled + (size_t)g * NODE_DIM;
    float s = p[lane] * Wfc[lane] + p[lane + 32] * Wfc[lane + 32];
#pragma unroll
    for (int off = 16; off > 0; off >>= 1)
        s += __shfl_down(s, off, 32);
    if (lane == 0) {
        float c = cnt[g];
        c = c < 1.0f ? 1.0f : c;
        out[g] = s / c + bfc[0];
    }
}

extern "C" void kernel_launch(void* const* d_in, const int* in_sizes, int n_in,
                              void* d_out, int out_size, void* d_ws, size_t ws_size,
                              hipStream_t stream) {
    const float* x     = (const float*)d_in[0];
    const int*   ei    = (const int*)d_in[1];    // [2, E]
    const int*   batch = (const int*)d_in[2];
    const float* W1    = (const float*)d_in[3];
    const float* b1    = (const float*)d_in[4];
    const float* W2    = (const float*)d_in[5];
    const float* b2    = (const float*)d_in[6];
    const float* Wfc   = (const float*)d_in[7];
    const float* bfc   = (const float*)d_in[8];
    float*       out   = (float*)d_out;

    const int N = in_sizes[0] / IN_DIM;     // 100000 (multiple of 16)
    const int E = in_sizes[1] / 2;          // 3.2M
    const int G = out_size;                 // 1024
    const int* src = ei;
    const int* dst = ei + E;

    // workspace layout (floats)
    float* ws     = (float*)d_ws;
    float* dinv   = ws;                                  // N      (also used as deg)
    float* bufA   = dinv + (size_t)N;                    // N*64   (xw / xw2)
    float* bufB   = bufA + (size_t)N * NODE_DIM;         // N*64   (agg / h1)
    float* pooled = bufB + (size_t)N * NODE_DIM;         // G*64
    float* cnt    = pooled + (size_t)G * NODE_DIM;       // G

    const long long ND = (long long)N * NODE_DIM;

    // ---- zero accumulators (capture-safe async memsets) ----
    hipMemsetAsync(dinv, 0, (size_t)N * sizeof(float), stream);
    hipMemsetAsync(pooled, 0, (size_t)(G * NODE_DIM + G) * sizeof(float), stream);
    hipMemsetAsync(bufB, 0, (size_t)ND * sizeof(float), stream);

    // ---- degree + symmetric norm ----
    k_degree<<<(E + 255) / 256, 256, 0, stream>>>(dst, dinv, E);
    k_dinv<<<(N + 255) / 256, 256, 0, stream>>>(dinv, N);

    // ---- layer 1 ----
    k_gemm_wmma<<<N / 16, 128, 0, stream>>>(x, W1, bufA, N, IN_DIM);
    {
        long long t = (long long)E * 16;
        k_scatter<<<(unsigned)((t + 255) / 256), 256, 0, stream>>>(src, dst, dinv, bufA, bufB, E);
    }
    k_self_bias_relu<<<(unsigned)((ND + 255) / 256), 256, 0, stream>>>(bufB, bufA, dinv, b1, N);

    // ---- layer 2 ----
    k_gemm_wmma<<<N / 16, 128, 0, stream>>>(bufB, W2, bufA, N, NODE_DIM);
    hipMemsetAsync(bufB, 0, (size_t)ND * sizeof(float), stream);   // ordered after gemm2
    {
        long long t = (long long)E * 16;
        k_scatter<<<(unsigned)((t + 255) / 256), 256, 0, stream>>>(src, dst, dinv, bufA, bufB, E);
    }
    k_relu_pool<<<(unsigned)((ND + 255) / 256), 256, 0, stream>>>(bufB, bufA, dinv, b2, batch,
                                                                  pooled, cnt, N);

    // ---- final FC over pooled means ----
    k_final<<<(G + 7) / 8, 256, 0, stream>>>(pooled, cnt, Wfc, bfc, out, G);
}